// SCTM_48713519071891
// MI455X (gfx1250) — compile-verified
//
#include <hip/hip_runtime.h>

typedef __attribute__((ext_vector_type(2))) float v2f;
typedef __attribute__((ext_vector_type(4))) float v4f;
typedef __attribute__((ext_vector_type(8))) float v8f;
typedef __attribute__((ext_vector_type(4))) int   v4i;

#define D_DIM 1024
#define B_DIM 8
#define T_DIM 512
#define NROWS (T_DIM * B_DIM)     // 4096
#define KH_SZ 33
#define CT_SZ 2
#define H_DIM (D_DIM + KH_SZ - 1) // 1056
#define KTAP  (CT_SZ * KH_SZ * KH_SZ) // 2178

// ---- CDNA5 async global->LDS copy (ASYNCcnt path), with safe fallback ----
#if __has_builtin(__builtin_amdgcn_global_load_async_to_lds_b128)
#define HAVE_ASYNC_LDS 1
#else
#define HAVE_ASYNC_LDS 0
#endif

// pointee-typed address-space pointers the builtin expects (param 0 is
// "v4i addrspace(1)*" per the round-2 diagnostic)
typedef __attribute__((address_space(1))) v4i* gbl_v4i_ptr;
typedef __attribute__((address_space(3))) v4i* lds_v4i_ptr;

__device__ __forceinline__ void cp16_async(float* dst_lds, const float* src_gbl) {
#if HAVE_ASYNC_LDS
  __builtin_amdgcn_global_load_async_to_lds_b128(
      (gbl_v4i_ptr)(unsigned long long)src_gbl,
      (lds_v4i_ptr)(unsigned)(unsigned long long)dst_lds,
      0, 0);
#else
  *(v4f*)dst_lds = *(const v4f*)src_gbl;
#endif
}

#if HAVE_ASYNC_LDS
#if __has_builtin(__builtin_amdgcn_s_wait_asynccnt)
#define WAIT_ASYNC(n) __builtin_amdgcn_s_wait_asynccnt((short)(n))
#else
#define WAIT_ASYNC(n) asm volatile("s_wait_asynccnt %0" ::"i"(n) : "memory")
#endif
#else
#define WAIT_ASYNC(n) do {} while (0)
#endif

// ---------------------------------------------------------------------------
// Phase 1: hypernet weight generation. 33x33x2 VALID conv over the template,
// producing 4 weight planes at once (lin, conv0, conv1, olin) so the template
// is streamed exactly once. Hyper-kernels (4*2178 floats = 35 KB) live in LDS.
// Output-channel count is 4 -> WMMA's 16-wide N would be 75% idle; VALU FMA
// is the right tool (~18 GFLOP, input L2-resident).
// ---------------------------------------------------------------------------
__global__ void hyper_weight_kernel(const float* __restrict__ tmpl,
                                    const float* __restrict__ hw_lin,
                                    const float* __restrict__ hw_conv,
                                    const float* __restrict__ hw_olin,
                                    const float* __restrict__ hb_lin,
                                    const float* __restrict__ hb_conv,
                                    const float* __restrict__ hb_olin,
                                    float* __restrict__ Wlin,
                                    float* __restrict__ Wc0,
                                    float* __restrict__ Wc1,
                                    float* __restrict__ Wolin) {
  __shared__ float shw[4 * KTAP];
  const int tid = threadIdx.x;
  for (int i = tid; i < KTAP; i += blockDim.x) {
    shw[i]            = hw_lin[i];
    shw[KTAP + i]     = hw_conv[i];
    shw[2 * KTAP + i] = hw_conv[KTAP + i];
    shw[3 * KTAP + i] = hw_olin[i];
  }
  __syncthreads();

  const int x = blockIdx.x * blockDim.x + tid;  // output col (i)
  const int y = blockIdx.y;                     // output row (o)
  float a0 = 0.f, a1 = 0.f, a2 = 0.f, a3 = 0.f;
#pragma unroll
  for (int ct = 0; ct < CT_SZ; ++ct) {
    const float* tp = tmpl + (size_t)ct * H_DIM * H_DIM + (size_t)y * H_DIM + x;
    const float* wp = shw + ct * KH_SZ * KH_SZ;
    for (int kh = 0; kh < KH_SZ; ++kh) {
      const float* trow = tp + kh * H_DIM;
      const float* wrow = wp + kh * KH_SZ;
#pragma unroll 11
      for (int kw = 0; kw < KH_SZ; ++kw) {
        const float tv = trow[kw];
        a0 = fmaf(tv, wrow[kw], a0);
        a1 = fmaf(tv, wrow[KTAP + kw], a1);
        a2 = fmaf(tv, wrow[2 * KTAP + kw], a2);
        a3 = fmaf(tv, wrow[3 * KTAP + kw], a3);
      }
    }
  }
  const size_t o = (size_t)y * D_DIM + x;
  Wlin[o]  = a0 + hb_lin[0];
  Wc0[o]   = a1 + hb_conv[0];
  Wc1[o]   = a2 + hb_conv[1];
  Wolin[o] = a3 + hb_olin[0];
}

// ---------------------------------------------------------------------------
// Phase 2: f32 WMMA GEMM with async double-buffered staging.
//   out[n,o] = bias[o] + A[n,:]·W0[o,:]  (+ DUAL: Ashift[n,:]·W1[o,:])
// Ashift row n = (n < B ? hidden : A[n-B]) is the K=2 causal conv stream.
// Block tile 64x256, Kc=32, ping-pong LDS buffers filled by
// global_load_async_to_lds_b128; wave tile 32x64 = 8 v8f accumulators.
// Fragment layout per CDNA5 ISA: A 16x4 f32 = 2 VGPRs (lanes 0-15: K=0,1;
// lanes 16-31: K=2,3); C/D: VGPR r <-> M=r (lanes 0-15) / M=r+8 (16-31).
// Per-thread async loads per tile: 10 (single) / 20 (dual); 2 tiles in
// flight max -> s_wait_asynccnt N before consuming a tile, 0 on the last.
// ---------------------------------------------------------------------------
template <bool DUAL>
__global__ void hyper_gemm_kernel(const float* __restrict__ A,
                                  const float* __restrict__ Ahid,
                                  const float* __restrict__ W0,
                                  const float* __restrict__ W1,
                                  const float* __restrict__ bias,
                                  float* __restrict__ out) {
  constexpr int BM = 64, BN = 256, KC = 32, SP = 36;  // SP: padded LDS stride
  constexpr int NT = D_DIM / KC;                      // 32 K-tiles
  constexpr int NLD = DUAL ? 20 : 10;                 // async loads/thread/tile

  __shared__ float Xs[2][BM * SP];
  __shared__ float Ws[2][BN * SP];
  __shared__ float Xs2[DUAL ? 2 : 1][DUAL ? BM * SP : 4];
  __shared__ float Ws2[DUAL ? 2 : 1][DUAL ? BN * SP : 4];

  const int tid  = threadIdx.x;
  const int lane = tid & 31;
  const int wid  = tid >> 5;
  const int wm   = wid & 1;   // wave M slot (0/1) -> rows 0/32
  const int wn   = wid >> 1;  // wave N slot (0..3) -> cols 0/64/128/192
  const int bm   = blockIdx.x * BM;
  const int bn   = blockIdx.y * BN;
  const int lh   = lane >> 4; // half-wave select
  const int lm   = lane & 15;

  // stage one KC-slice of all operand tiles into LDS buffer `buf`
  auto issue_tile = [&](int kt, int buf) {
    for (int idx = tid; idx < BM * (KC / 4); idx += 256) {
      const int r = idx >> 3;
      const int c = (idx & 7) << 2;
      cp16_async(&Xs[buf][r * SP + c], A + (size_t)(bm + r) * D_DIM + kt + c);
      if (DUAL) {
        const int rg = bm + r;
        const float* src = (rg < B_DIM)
                               ? (Ahid + kt + c)
                               : (A + (size_t)(rg - B_DIM) * D_DIM + kt + c);
        cp16_async(&Xs2[buf][r * SP + c], src);
      }
    }
    for (int idx = tid; idx < BN * (KC / 4); idx += 256) {
      const int r = idx >> 3;
      const int c = (idx & 7) << 2;
      cp16_async(&Ws[buf][r * SP + c], W0 + (size_t)(bn + r) * D_DIM + kt + c);
      if (DUAL)
        cp16_async(&Ws2[buf][r * SP + c], W1 + (size_t)(bn + r) * D_DIM + kt + c);
    }
  };

  v8f acc[2][4];
#pragma unroll
  for (int mi = 0; mi < 2; ++mi)
#pragma unroll
    for (int ni = 0; ni < 4; ++ni) acc[mi][ni] = {};

  issue_tile(0, 0);
  issue_tile(KC, 1);

  for (int kt_i = 0; kt_i < NT; ++kt_i) {
    if (kt_i + 1 < NT) { WAIT_ASYNC(NLD); } else { WAIT_ASYNC(0); }
    __syncthreads();  // all waves' tile kt_i LDS writes visible

    const int buf = kt_i & 1;
    const float* Xb  = Xs[buf];
    const float* Wb  = Ws[buf];
    const float* Xb2 = Xs2[DUAL ? buf : 0];
    const float* Wb2 = Ws2[DUAL ? buf : 0];

#pragma unroll
    for (int kk = 0; kk < KC; kk += 4) {
      const int ko = kk + (lh << 1);  // lanes 16-31 take K=2,3 of the step
      v2f a0 = *(const v2f*)(&Xb[(wm * 32 + lm) * SP + ko]);
      v2f a1 = *(const v2f*)(&Xb[(wm * 32 + 16 + lm) * SP + ko]);
      v2f b0 = *(const v2f*)(&Wb[(wn * 64 + lm) * SP + ko]);
      v2f b1 = *(const v2f*)(&Wb[(wn * 64 + 16 + lm) * SP + ko]);
      v2f b2 = *(const v2f*)(&Wb[(wn * 64 + 32 + lm) * SP + ko]);
      v2f b3 = *(const v2f*)(&Wb[(wn * 64 + 48 + lm) * SP + ko]);
      acc[0][0] = __builtin_amdgcn_wmma_f32_16x16x4_f32(false, a0, false, b0, (short)0, acc[0][0], false, false);
      acc[0][1] = __builtin_amdgcn_wmma_f32_16x16x4_f32(false, a0, false, b1, (short)0, acc[0][1], false, false);
      acc[0][2] = __builtin_amdgcn_wmma_f32_16x16x4_f32(false, a0, false, b2, (short)0, acc[0][2], false, false);
      acc[0][3] = __builtin_amdgcn_wmma_f32_16x16x4_f32(false, a0, false, b3, (short)0, acc[0][3], false, false);
      acc[1][0] = __builtin_amdgcn_wmma_f32_16x16x4_f32(false, a1, false, b0, (short)0, acc[1][0], false, false);
      acc[1][1] = __builtin_amdgcn_wmma_f32_16x16x4_f32(false, a1, false, b1, (short)0, acc[1][1], false, false);
      acc[1][2] = __builtin_amdgcn_wmma_f32_16x16x4_f32(false, a1, false, b2, (short)0, acc[1][2], false, false);
      acc[1][3] = __builtin_amdgcn_wmma_f32_16x16x4_f32(false, a1, false, b3, (short)0, acc[1][3], false, false);
      if (DUAL) {
        v2f c0 = *(const v2f*)(&Xb2[(wm * 32 + lm) * SP + ko]);
        v2f c1 = *(const v2f*)(&Xb2[(wm * 32 + 16 + lm) * SP + ko]);
        v2f d0 = *(const v2f*)(&Wb2[(wn * 64 + lm) * SP + ko]);
        v2f d1 = *(const v2f*)(&Wb2[(wn * 64 + 16 + lm) * SP + ko]);
        v2f d2 = *(const v2f*)(&Wb2[(wn * 64 + 32 + lm) * SP + ko]);
        v2f d3 = *(const v2f*)(&Wb2[(wn * 64 + 48 + lm) * SP + ko]);
        acc[0][0] = __builtin_amdgcn_wmma_f32_16x16x4_f32(false, c0, false, d0, (short)0, acc[0][0], false, false);
        acc[0][1] = __builtin_amdgcn_wmma_f32_16x16x4_f32(false, c0, false, d1, (short)0, acc[0][1], false, false);
        acc[0][2] = __builtin_amdgcn_wmma_f32_16x16x4_f32(false, c0, false, d2, (short)0, acc[0][2], false, false);
        acc[0][3] = __builtin_amdgcn_wmma_f32_16x16x4_f32(false, c0, false, d3, (short)0, acc[0][3], false, false);
        acc[1][0] = __builtin_amdgcn_wmma_f32_16x16x4_f32(false, c1, false, d0, (short)0, acc[1][0], false, false);
        acc[1][1] = __builtin_amdgcn_wmma_f32_16x16x4_f32(false, c1, false, d1, (short)0, acc[1][1], false, false);
        acc[1][2] = __builtin_amdgcn_wmma_f32_16x16x4_f32(false, c1, false, d2, (short)0, acc[1][2], false, false);
        acc[1][3] = __builtin_amdgcn_wmma_f32_16x16x4_f32(false, c1, false, d3, (short)0, acc[1][3], false, false);
      }
    }
    __syncthreads();  // everyone done reading buf before it is refilled
    if (kt_i + 2 < NT) issue_tile((kt_i + 2) * KC, buf);
  }

  // epilogue: bias add + store (C layout: VGPR r <-> M = r + 8*lh, N = lm)
#pragma unroll
  for (int ni = 0; ni < 4; ++ni) {
    const int col = bn + wn * 64 + ni * 16 + lm;
    const float bv = bias[col];
#pragma unroll
    for (int mi = 0; mi < 2; ++mi) {
      const int rbase = bm + wm * 32 + mi * 16 + (lh << 3);
#pragma unroll
      for (int r = 0; r < 8; ++r) {
        out[(size_t)(rbase + r) * D_DIM + col] = acc[mi][ni][r] + bv;
      }
    }
  }
}

// ---------------------------------------------------------------------------
// Phase 3: sequential scan over T (8192 independent (b,o) lanes).
//   h = c*lv + cv ; c = h*cv.  Writes h_all straight into d_out's new_h slot.
// ---------------------------------------------------------------------------
__global__ void scan_kernel(const float* __restrict__ lv,
                            const float* __restrict__ cv,
                            const float* __restrict__ cell,
                            float* __restrict__ h_all,
                            float* __restrict__ new_c) {
  const int g = blockIdx.x * blockDim.x + threadIdx.x;  // 0..B*D-1
  const int o = g & (D_DIM - 1);
  float c = cell[o];  // cell_state [1,D] broadcast over b
  size_t idx = (size_t)g;
  for (int t = 0; t < T_DIM; ++t) {
    const float l = lv[idx];
    const float v = cv[idx];
    const float h = fmaf(c, l, v);
    h_all[idx] = h;
    c = h * v;
    idx += (size_t)B_DIM * D_DIM;
  }
  new_c[g] = c;
}

// ---------------------------------------------------------------------------
extern "C" void kernel_launch(void* const* d_in, const int* in_sizes, int n_in,
                              void* d_out, int out_size, void* d_ws, size_t ws_size,
                              hipStream_t stream) {
  (void)in_sizes; (void)n_in; (void)out_size; (void)ws_size;
  const float* vec     = (const float*)d_in[0];
  const float* tmpl    = (const float*)d_in[1];
  const float* hid     = (const float*)d_in[2];
  const float* cell    = (const float*)d_in[3];
  const float* hw_lin  = (const float*)d_in[4];
  const float* hb_lin  = (const float*)d_in[5];
  const float* b_lin   = (const float*)d_in[6];
  const float* hw_conv = (const float*)d_in[7];
  const float* hb_conv = (const float*)d_in[8];
  const float* b_conv  = (const float*)d_in[9];
  const float* hw_olin = (const float*)d_in[10];
  const float* hb_olin = (const float*)d_in[11];
  const float* b_olin  = (const float*)d_in[12];

  const size_t WSZ = (size_t)D_DIM * D_DIM;   // 1M floats
  const size_t ASZ = (size_t)NROWS * D_DIM;   // 4M floats
  float* ws    = (float*)d_ws;
  float* Wlin  = ws;
  float* Wc0   = ws + WSZ;
  float* Wc1   = ws + 2 * WSZ;
  float* Wolin = ws + 3 * WSZ;
  float* lv    = ws + 4 * WSZ;
  float* cv    = lv + ASZ;                    // total 12M floats = 48 MB

  float* ret  = (float*)d_out;                // [NROWS, D]
  float* newh = ret + ASZ;                    // [NROWS, D]
  float* newc = newh + ASZ;                   // [B, D]

  // 1) generate all four hypernet weight planes
  hyper_weight_kernel<<<dim3(D_DIM / 256, D_DIM), 256, 0, stream>>>(
      tmpl, hw_lin, hw_conv, hw_olin, hb_lin, hb_conv, hb_olin,
      Wlin, Wc0, Wc1, Wolin);

  // 2) lv = vec · Wlin^T + b_lin
  hyper_gemm_kernel<false><<<dim3(NROWS / 64, D_DIM / 256), 256, 0, stream>>>(
      vec, nullptr, Wlin, nullptr, b_lin, lv);

  // 3) cv = b_conv + vec·Wc1^T + shift(vec, hidden)·Wc0^T   (fused dual GEMM)
  hyper_gemm_kernel<true><<<dim3(NROWS / 64, D_DIM / 256), 256, 0, stream>>>(
      vec, hid, Wc1, Wc0, b_conv, cv);

  // 4) recurrence scan -> new_h (into d_out) and new_c
  scan_kernel<<<dim3((B_DIM * D_DIM) / 256), 256, 0, stream>>>(
      lv, cv, cell, newh, newc);

  // 5) ret = h_all · Wolin^T + b_olin
  hyper_gemm_kernel<false><<<dim3(NROWS / 64, D_DIM / 256), 256, 0, stream>>>(
      newh, nullptr, Wolin, nullptr, b_olin, ret);
}